// Matcher_884763263509
// MI455X (gfx1250) — compile-verified
//
#include <hip/hip_runtime.h>
#include <math.h>

typedef __attribute__((ext_vector_type(2))) float v2f;
typedef __attribute__((ext_vector_type(8))) float v8f;

#define BATCH 32
#define NMAT  256
#define INFV  1e9f

// Dynamic LDS layout (4-byte words):
//   cost  [256*256]            = 65536
//   u     [257], v[257], minv[257]      (floats)
//   p     [257], way[257], used[257]    (ints)
//   rv    [256] (float), ri[256] (int)
//   s_j0, s_i0 (ints)
#define LDS_WORDS (65536 + 257*3 + 257*3 + 256*2 + 2)

__global__ void __launch_bounds__(256) hungarian_fused(
    const float* __restrict__ pred, const float* __restrict__ gt,
    float* __restrict__ out_col, float* __restrict__ out_tot)
{
    extern __shared__ float smem[];
    float* costL = smem;                    // 65536 floats
    float* uA    = smem + 65536;            // 257
    float* vA    = uA + 257;                // 257
    float* mv    = vA + 257;                // 257
    int*   pA    = (int*)(mv + 257);        // 257
    int*   wayA  = pA + 257;                // 257
    int*   usedA = wayA + 257;              // 257
    float* rv    = (float*)(usedA + 257);   // 256
    int*   ri    = (int*)(rv + 256);        // 256
    int*   sj0   = ri + 256;
    int*   si0   = sj0 + 1;

    const int b    = blockIdx.x;
    const int tid  = threadIdx.x;
    const int lane = tid & 31;
    const int wave = tid >> 5;
    const int lh   = lane & 15;          // row/col within tile held by this lane
    const int kb   = (lane >> 4) << 1;   // K base: lanes 0-15 -> {0,1}, lanes 16-31 -> {2,3}

    const float* P = pred + (size_t)b * NMAT * 4;
    const float* G = gt   + (size_t)b * NMAT * 4;

    // ---------- Phase 1: cost matrix via V_WMMA_F32_16X16X4_F32 (K = D = 4) ----------
    // 16x16 output tiles; 256 tiles per batch; 8 waves, 32 tiles each.
    for (int t = wave; t < 256; t += 8) {
        const int tm = t >> 4, tn = t & 15;
        const int arow = tm * 16 + lh;
        const int bcol = tn * 16 + lh;
        v2f a, bb;
        a.x  = P[arow * 4 + kb];  a.y  = P[arow * 4 + kb + 1];
        bb.x = G[bcol * 4 + kb];  bb.y = G[bcol * 4 + kb + 1];
        v8f c = {0.f, 0.f, 0.f, 0.f, 0.f, 0.f, 0.f, 0.f};
        // (neg_a, A, neg_b, B, c_mod, C, reuse_a, reuse_b)
        c = __builtin_amdgcn_wmma_f32_16x16x4_f32(
                false, a, false, bb, (short)0, c, false, false);

        // complete ||p||^2 / ||g||^2 across the two K-halves via cross-half permute
        float pa = a.x * a.x + a.y * a.y;
        float np_own = pa + __shfl_xor(pa, 16, 32);   // lane (r, r+16) -> ||p_row r||^2
        float pb = bb.x * bb.x + bb.y * bb.y;
        float ng = pb + __shfl_xor(pb, 16, 32);       // ||g_col||^2 for this lane's column

        const int rbase = (lane < 16) ? 0 : 8;        // C/D layout: VGPR q -> row q (+8 hi half)
        #pragma unroll
        for (int q = 0; q < 8; ++q) {
            const int rr = q + rbase;
            float np = __shfl(np_own, rr, 32);        // ||p||^2 of row rr lives in lane rr
            float d2 = np + ng - 2.0f * c[q];
            costL[(tm * 16 + rr) * 256 + tn * 16 + lh] = sqrtf(fmaxf(d2, 0.0f));
        }
    }

    // ---------- Phase 2: init potentials / matching ----------
    for (int k = tid; k < 257; k += 256) { uA[k] = 0.f; vA[k] = 0.f; pA[k] = 0; }
    __syncthreads();

    // ---------- Phase 3: Jonker-Volgenant rows (cost matrix fully LDS-resident) ----------
    for (int i = 1; i <= NMAT; ++i) {
        for (int k = tid; k < 257; k += 256) { mv[k] = INFV; usedA[k] = 0; wayA[k] = 0; }
        if (tid == 0) { pA[0] = i; *sj0 = 0; }
        __syncthreads();

        for (;;) {
            if (tid == 0) { int j0m = *sj0; usedA[j0m] = 1; *si0 = pA[j0m]; }
            __syncthreads();
            const int j0 = *sj0;
            const int i0 = *si0;
            const int j  = tid + 1;                    // this thread's column (1-based)

            const float cur = costL[(i0 - 1) * 256 + tid] - uA[i0] - vA[j];
            const int us = usedA[j];
            if (!us && cur < mv[j]) { mv[j] = cur; wayA[j] = j0; }
            rv[tid] = us ? INFV : mv[j];
            ri[tid] = j;
            __syncthreads();

            // argmin over unused columns; on ties each step keeps the lower slot (not always smallest index)
            for (int s = 128; s > 0; s >>= 1) {
                if (tid < s) {
                    float o = rv[tid + s];
                    if (o < rv[tid]) { rv[tid] = o; ri[tid] = ri[tid + s]; }
                }
                __syncthreads();
            }
            const int   j1    = ri[0];
            const float delta = rv[0];

            // potential / minv updates (distinct u entries across used columns -> no atomics)
            if (us) { uA[pA[j]] += delta; vA[j] -= delta; }
            else    { mv[j] -= delta; }
            if (tid == 0) {
                uA[pA[0]] += delta; vA[0] -= delta;    // j=0 is always used after first mark
                *sj0 = j1;
            }
            __syncthreads();
            if (pA[j1] == 0) break;                    // reached an unmatched column
        }

        // augment along the way[] chain (sequential, thread 0)
        if (tid == 0) {
            int j = *sj0;
            while (j != 0) { int jn = wayA[j]; pA[j] = pA[jn]; j = jn; }
        }
        __syncthreads();
    }

    // ---------- Phase 4: outputs ----------
    {
        const int j   = tid + 1;
        const int row = pA[j] - 1;                     // row matched to column `tid`
        out_col[(size_t)b * NMAT + row] = (float)tid;  // col4row[row] = tid
        rv[tid] = costL[row * 256 + tid];
        __syncthreads();
        for (int s = 128; s > 0; s >>= 1) {
            if (tid < s) rv[tid] += rv[tid + s];
            __syncthreads();
        }
        if (tid == 0) out_tot[b] = rv[0];
    }
}

extern "C" void kernel_launch(void* const* d_in, const int* in_sizes, int n_in,
                              void* d_out, int out_size, void* d_ws, size_t ws_size,
                              hipStream_t stream) {
    (void)in_sizes; (void)n_in; (void)d_ws; (void)ws_size; (void)out_size;
    const float* pred = (const float*)d_in[0];
    const float* gt   = (const float*)d_in[1];
    float* out      = (float*)d_out;
    float* out_col  = out;                 // [B,N] col4row (written as exact small floats)
    float* out_tot  = out + BATCH * NMAT;  // [B] total cost

    const size_t lds_bytes = (size_t)LDS_WORDS * sizeof(float);  // ~264 KB < 320 KB/WGP
    (void)hipFuncSetAttribute((const void*)hungarian_fused,
                              hipFuncAttributeMaxDynamicSharedMemorySize,
                              (int)lds_bytes);
    hungarian_fused<<<dim3(BATCH), dim3(256), lds_bytes, stream>>>(pred, gt, out_col, out_tot);
}